// GNNVAE_28621662060780
// MI455X (gfx1250) — compile-verified
//
#include <hip/hip_runtime.h>

typedef float v2f __attribute__((ext_vector_type(2)));
typedef float v8f __attribute__((ext_vector_type(8)));

// ---------------------------------------------------------------------------
// Degree / normalization kernels
// ---------------------------------------------------------------------------
__global__ __launch_bounds__(256) void init_deg_kernel(float* __restrict__ deg, int n) {
    int i = blockIdx.x * blockDim.x + threadIdx.x;
    if (i < n) deg[i] = 1.0f;   // self-loop contributes 1 to every node's degree
}

__global__ __launch_bounds__(256) void deg_scatter_kernel(const int* __restrict__ dst,
                                                          float* __restrict__ deg, int e) {
    int i = blockIdx.x * blockDim.x + threadIdx.x;
    if (i < e) atomicAdd(&deg[dst[i]], 1.0f);
}

__global__ __launch_bounds__(256) void dinv_kernel(const float* __restrict__ deg,
                                                   float* __restrict__ dinv, int n) {
    int i = blockIdx.x * blockDim.x + threadIdx.x;
    if (i < n) dinv[i] = rsqrtf(deg[i]);   // deg >= 1 always (self-loops)
}

// ---------------------------------------------------------------------------
// fp32 WMMA GEMM:  C[M x N] = A[M x K] * B[K x N]
// Block = 128 threads (4 waves). Each block computes 64 rows x N cols.
// LDS: 64 x K A-tile (32KB for K=128) + K x 64 B-slice (32KB) = 64KB total.
// Uses V_WMMA_F32_16X16X4_F32; accumulation over K in steps of 4.
// ---------------------------------------------------------------------------
template <int K, int N>
__global__ __launch_bounds__(128) void gemm_wmma_kernel(const float* __restrict__ A,
                                                        const float* __restrict__ B,
                                                        float* __restrict__ C, int M) {
    __shared__ float Alds[64 * K];
    __shared__ float Blds[K * 64];

    const int tid     = threadIdx.x;           // 0..127
    const int wave    = tid >> 5;              // 0..3
    const int lane    = tid & 31;
    const int lhalf   = lane >> 4;             // 0 for lanes 0-15, 1 for 16-31
    const int l15     = lane & 15;
    const int rowBase = blockIdx.x * 64;

    // Cooperative A-tile load (zero-pad rows past M; keeps EXEC full at WMMA)
    for (int i = tid; i < 64 * K; i += 128) {
        int r = rowBase + i / K;
        Alds[i] = (r < M) ? A[(size_t)r * K + (i % K)] : 0.0f;
    }

    const int mRow = wave * 16 + l15;          // A-frag row within block tile

    for (int nb = 0; nb < N; nb += 64) {       // N-slice loop (1 or 2 iters)
        __syncthreads();                       // A ready / Blds safe to overwrite
        for (int i = tid; i < K * 64; i += 128) {
            int kk = i >> 6, nn = i & 63;
            Blds[i] = B[(size_t)kk * N + nb + nn];
        }
        __syncthreads();

        for (int nt = 0; nt < 4; ++nt) {       // four 16-wide output tiles
            const int n = nt * 16 + l15;       // B/C column within slice
            v8f acc = {};
            for (int k = 0; k < K; k += 4) {
                const int ka = k + lhalf * 2;  // ISA 32-bit A/B lane->K mapping
                v2f a, b;
                a.x = Alds[mRow * K + ka];
                a.y = Alds[mRow * K + ka + 1];
                b.x = Blds[ka * 64 + n];
                b.y = Blds[(ka + 1) * 64 + n];
                acc = __builtin_amdgcn_wmma_f32_16x16x4_f32(
                    /*neg_a=*/false, a, /*neg_b=*/false, b,
                    /*c_mod=*/(short)0, acc, /*reuse_a=*/false, /*reuse_b=*/false);
            }
            // C/D layout: VGPR g holds M = g + 8*lhalf, N = l15
            const int row0 = rowBase + wave * 16 + lhalf * 8;
            const int col  = nb + nt * 16 + l15;
#pragma unroll
            for (int g = 0; g < 8; ++g) {
                int r = row0 + g;
                if (r < M) C[(size_t)r * N + col] = acc[g];
            }
        }
    }
}

// ---------------------------------------------------------------------------
// Aggregation: AGG = sum over (src->dst) edges of H[src]*dinv[src]*dinv[dst],
// plus self-loop term H[i]*dinv[i]^2.  Then bias + ReLU.
// ---------------------------------------------------------------------------
template <int F>
__global__ __launch_bounds__(256) void self_init_kernel(const float* __restrict__ H,
                                                        const float* __restrict__ dinv,
                                                        float* __restrict__ AGG, int n) {
    int t = blockIdx.x * blockDim.x + threadIdx.x;       // one float4 per thread
    int total = n * (F / 4);
    if (t >= total) return;
    int node = t / (F / 4);
    int fg   = t % (F / 4);
    float w  = dinv[node] * dinv[node];
    const float4* h = (const float4*)(H + (size_t)node * F);
    float4 v = h[fg];
    float4* o = (float4*)(AGG + (size_t)node * F);
    o[fg] = make_float4(v.x * w, v.y * w, v.z * w, v.w * w);
}

template <int F>
__global__ __launch_bounds__(256) void edge_scatter_kernel(const int* __restrict__ src,
                                                           const int* __restrict__ dst,
                                                           const float* __restrict__ dinv,
                                                           const float* __restrict__ H,
                                                           float* __restrict__ AGG, int e) {
    int t = blockIdx.x * blockDim.x + threadIdx.x;       // (edge, float4-group)
    int total = e * (F / 4);
    if (t >= total) return;
    int ei = t / (F / 4);
    int fg = t % (F / 4);
    int s = src[ei], d = dst[ei];
    float w = dinv[s] * dinv[d];
    const float4* h = (const float4*)(H + (size_t)s * F);
    float4 v = h[fg];
    float* o = AGG + (size_t)d * F + fg * 4;
    atomicAdd(o + 0, v.x * w);
    atomicAdd(o + 1, v.y * w);
    atomicAdd(o + 2, v.z * w);
    atomicAdd(o + 3, v.w * w);
}

template <int F>
__global__ __launch_bounds__(256) void bias_relu_kernel(float* __restrict__ Z,
                                                        const float* __restrict__ bias,
                                                        int n) {
    int t = blockIdx.x * blockDim.x + threadIdx.x;
    int total = n * F;
    if (t >= total) return;
    int f = t % F;
    float v = Z[t] + bias[f];
    Z[t] = v > 0.0f ? v : 0.0f;
}

// ---------------------------------------------------------------------------
// Decoder: one wave32 per pair; each lane covers 2 of 64 dims (float2),
// tree-reduce across the wave with shfl_xor (wave32 lane ops).
// ---------------------------------------------------------------------------
__global__ __launch_bounds__(256) void decoder_kernel(const float* __restrict__ Z,
                                                      const int* __restrict__ ia,
                                                      const int* __restrict__ ib,
                                                      float* __restrict__ out, int npairs) {
    int pair = blockIdx.x * 8 + (threadIdx.x >> 5);
    int lane = threadIdx.x & 31;
    if (pair >= npairs) return;                 // uniform per wave
    int a = ia[pair], b = ib[pair];
    const float2* za = (const float2*)(Z + (size_t)a * 64);
    const float2* zb = (const float2*)(Z + (size_t)b * 64);
    float2 x = za[lane], y = zb[lane];
    float p = x.x * y.x + x.y * y.y;
#pragma unroll
    for (int off = 16; off > 0; off >>= 1) p += __shfl_xor(p, off, 32);
    if (lane == 0) out[pair] = p;
}

// ---------------------------------------------------------------------------
// Host-side launch
// ---------------------------------------------------------------------------
extern "C" void kernel_launch(void* const* d_in, const int* in_sizes, int n_in,
                              void* d_out, int out_size, void* d_ws, size_t ws_size,
                              hipStream_t stream) {
    const float* x     = (const float*)d_in[0];
    const float* W_in  = (const float*)d_in[1];
    const float* b_in  = (const float*)d_in[2];
    const float* W_out = (const float*)d_in[3];
    const float* b_out = (const float*)d_in[4];
    const int*   ei    = (const int*)d_in[5];   // [2, E] row-major: src then dst
    const int*   eli   = (const int*)d_in[6];   // [2, NDEC]

    const int N    = in_sizes[0] / 128;   // 50000 nodes
    const int E    = in_sizes[5] / 2;     // 800000 edges
    const int NDEC = in_sizes[6] / 2;     // 1000000 pairs
    const int* src = ei;
    const int* dst = ei + E;
    const int* ia  = eli;
    const int* ib  = eli + NDEC;
    float* logits  = (float*)d_out;

    // Workspace layout (256B aligned slices)
    char* base = (char*)d_ws;
    size_t off = 0;
    auto take = [&](size_t bytes) -> float* {
        float* p = (float*)(base + off);
        off = (off + bytes + 255) & ~(size_t)255;
        return p;
    };
    float* deg  = take((size_t)N * 4);
    float* dinv = take((size_t)N * 4);
    float* H1   = take((size_t)N * 128 * 4);
    float* A1   = take((size_t)N * 128 * 4);   // becomes Z1 after bias+ReLU
    float* H2   = take((size_t)N * 64 * 4);
    float* A2   = take((size_t)N * 64 * 4);    // becomes Z2 after bias+ReLU
    (void)ws_size;

    const int TB = 256;
    // 1) degrees + normalization
    init_deg_kernel<<<(N + TB - 1) / TB, TB, 0, stream>>>(deg, N);
    deg_scatter_kernel<<<(E + TB - 1) / TB, TB, 0, stream>>>(dst, deg, E);
    dinv_kernel<<<(N + TB - 1) / TB, TB, 0, stream>>>(deg, dinv, N);

    // 2) layer 1: H1 = x @ W_in  (WMMA fp32), then normalized aggregation
    gemm_wmma_kernel<128, 128><<<(N + 63) / 64, 128, 0, stream>>>(x, W_in, H1, N);
    {
        int total = N * 32;
        self_init_kernel<128><<<(total + TB - 1) / TB, TB, 0, stream>>>(H1, dinv, A1, N);
        int etotal = E * 32;
        edge_scatter_kernel<128><<<(etotal + TB - 1) / TB, TB, 0, stream>>>(src, dst, dinv, H1, A1, E);
        int ztotal = N * 128;
        bias_relu_kernel<128><<<(ztotal + TB - 1) / TB, TB, 0, stream>>>(A1, b_in, N);
    }

    // 3) layer 2: H2 = Z1 @ W_out (WMMA fp32), then normalized aggregation
    gemm_wmma_kernel<128, 64><<<(N + 63) / 64, 128, 0, stream>>>(A1, W_out, H2, N);
    {
        int total = N * 16;
        self_init_kernel<64><<<(total + TB - 1) / TB, TB, 0, stream>>>(H2, dinv, A2, N);
        int etotal = E * 16;
        edge_scatter_kernel<64><<<(etotal + TB - 1) / TB, TB, 0, stream>>>(src, dst, dinv, H2, A2, E);
        int ztotal = N * 64;
        bias_relu_kernel<64><<<(ztotal + TB - 1) / TB, TB, 0, stream>>>(A2, b_out, N);
    }

    // 4) decoder: logits = rowwise dot(Z2[a], Z2[b])
    decoder_kernel<<<(NDEC + 7) / 8, 256, 0, stream>>>(A2, ia, ib, logits, NDEC);
}